// QLSTM_65481071408219
// MI455X (gfx1250) — compile-verified
//
#include <hip/hip_runtime.h>
#include <math.h>

typedef __attribute__((ext_vector_type(2))) float v2f;
typedef __attribute__((ext_vector_type(8))) float v8f;

#define T_STEPS 512
#define BATCH   256
#define DIM     128
#define HID     128

// LDS row strides chosen for conflict-free WMMA-fragment access (64 banks):
#define HSTR 132   // hbuf/cbuf/ibuf/sWh: bank = (4*lm + k0) % 64 -> 64 distinct
#define USTR 12    // sU:                 bank = (12*lm + k0) % 64 -> distinct
#define ZSTR 36    // zbuf/qbuf:          bank = (36*lm + k0) % 64 -> distinct

__device__ __forceinline__ float sigmoidf_(float x) {
    return 1.0f / (1.0f + __expf(-x));
}
__device__ __forceinline__ float tanhf_(float x) {
    // 1 - 2/(e^{2x}+1): exact limits at +/-inf, one hardware exp
    return 1.0f - 2.0f / (__expf(2.0f * x) + 1.0f);
}

// ---------------------------------------------------------------------------
// Kernel 1: zx[t*B+b, n] = x[t,b,:] @ Wx_{gate(n)}[q(n),:].T + bias[n]
// M-tile = 16 rows/wave, N = 32 (2 tiles fused), K = 128, split-K x2:
// 4 independent WMMA accumulator chains.
// ---------------------------------------------------------------------------
__global__ __launch_bounds__(128) void qlstm_zx_kernel(
    const float* __restrict__ X,                       // (T*B, 128)
    const float* __restrict__ Wf, const float* __restrict__ bf,
    const float* __restrict__ Wi, const float* __restrict__ bi,
    const float* __restrict__ Wg, const float* __restrict__ bg,
    const float* __restrict__ Wo, const float* __restrict__ bo,
    float* __restrict__ zx)                            // (T*B, 32)
{
    __shared__ float sW[32 * HSTR];  // sW[n][k] (padded) = x-part of W
    __shared__ float sB[32];

    const int tid  = threadIdx.x;      // 0..127
    const int lane = tid & 31;
    const int wave = tid >> 5;         // 0..3
    const int half = lane >> 4;        // 0/1
    const int lm   = lane & 15;

    for (int g = 0; g < 4; ++g) {
        const float* W = (g == 0) ? Wf : (g == 1) ? Wi : (g == 2) ? Wg : Wo;
        const float* b = (g == 0) ? bf : (g == 1) ? bi : (g == 2) ? bg : bo;
        for (int i = tid; i < 8 * 128; i += 128) {
            int q = i >> 7, k = i & 127;
            sW[(g * 8 + q) * HSTR + k] = W[q * 256 + k];   // first 128 cols = x part
        }
        if (tid < 8) sB[g * 8 + tid] = b[tid];
    }
    __syncthreads();

    const int tile    = blockIdx.x * 4 + wave;   // M-tile index over T*B/16
    const int rowbase = tile * 16;
    const float* Xrow = X + (size_t)(rowbase + lm) * DIM;

    float bias0 = sB[lm], bias1 = sB[16 + lm];
    v8f c0, c1, d0 = {}, d1 = {};
    #pragma unroll
    for (int r = 0; r < 8; ++r) { c0[r] = bias0; c1[r] = bias1; }

    #pragma unroll 4
    for (int kk = 0; kk < 128; kk += 8) {
        int ka = kk + 2 * half;
        int kb = kk + 4 + 2 * half;
        v2f a0  = *(const v2f*)(Xrow + ka);
        v2f a1  = *(const v2f*)(Xrow + kb);
        v2f b00 = *(const v2f*)(sW + lm * HSTR + ka);
        v2f b10 = *(const v2f*)(sW + (16 + lm) * HSTR + ka);
        v2f b01 = *(const v2f*)(sW + lm * HSTR + kb);
        v2f b11 = *(const v2f*)(sW + (16 + lm) * HSTR + kb);
        c0 = __builtin_amdgcn_wmma_f32_16x16x4_f32(false, a0, false, b00, (short)0, c0, false, false);
        c1 = __builtin_amdgcn_wmma_f32_16x16x4_f32(false, a0, false, b10, (short)0, c1, false, false);
        d0 = __builtin_amdgcn_wmma_f32_16x16x4_f32(false, a1, false, b01, (short)0, d0, false, false);
        d1 = __builtin_amdgcn_wmma_f32_16x16x4_f32(false, a1, false, b11, (short)0, d1, false, false);
    }
    c0 = c0 + d0;
    c1 = c1 + d1;

    #pragma unroll
    for (int r = 0; r < 8; ++r) {
        size_t row = (size_t)(rowbase + r + 8 * half) * 32;
        zx[row + lm]      = c0[r];
        zx[row + 16 + lm] = c1[r];
    }
}

// ---------------------------------------------------------------------------
// Kernel 2: sequential scan.  16 single-wave workgroups, one 16-row batch
// chunk each, zero inter-wave sync.  Per step: zh GEMM (64 WMMA in 4
// independent chains), qgate, 4 x U GEMM (64 WMMA, 4-wide ILP), fused LSTM.
// ---------------------------------------------------------------------------
__global__ __launch_bounds__(32) void qlstm_scan_kernel(
    const float* __restrict__ zx,                      // (T*B, 32)
    const float* __restrict__ Wf, const float* __restrict__ thf,
    const float* __restrict__ Uf, const float* __restrict__ cf,
    const float* __restrict__ Wi, const float* __restrict__ thi,
    const float* __restrict__ Ui, const float* __restrict__ ci,
    const float* __restrict__ Wg, const float* __restrict__ thg,
    const float* __restrict__ Ug, const float* __restrict__ cg,
    const float* __restrict__ Wo, const float* __restrict__ tho,
    const float* __restrict__ Uo, const float* __restrict__ co,
    float* __restrict__ outs,                          // (T, B, H)
    float* __restrict__ hx, float* __restrict__ cx)    // (B, H) each
{
    __shared__ float sWh[32 * HSTR];      // h-part of W, padded rows
    __shared__ float sU[4 * 128 * USTR];  // sU[g][h][q], padded rows
    __shared__ float sTh[32];
    __shared__ float sCb[4 * 128];
    __shared__ float hbuf[16 * HSTR];
    __shared__ float cbuf[16 * HSTR];
    __shared__ float ibuf[16 * HSTR];
    __shared__ float zbuf[16 * ZSTR];
    __shared__ float qbuf[16 * ZSTR];

    const int lane = threadIdx.x;      // 32-thread (1-wave) block
    const int half = lane >> 4;
    const int lm   = lane & 15;

    for (int g = 0; g < 4; ++g) {
        const float* W  = (g == 0) ? Wf  : (g == 1) ? Wi  : (g == 2) ? Wg  : Wo;
        const float* U  = (g == 0) ? Uf  : (g == 1) ? Ui  : (g == 2) ? Ug  : Uo;
        const float* th = (g == 0) ? thf : (g == 1) ? thi : (g == 2) ? thg : tho;
        const float* cb = (g == 0) ? cf  : (g == 1) ? ci  : (g == 2) ? cg  : co;
        for (int i = lane; i < 8 * 128; i += 32) {
            int q = i >> 7, k = i & 127;
            sWh[(g * 8 + q) * HSTR + k] = W[q * 256 + 128 + k];   // h part
        }
        for (int i = lane; i < 128 * 8; i += 32) {
            int h = i >> 3, q = i & 7;
            sU[(g * 128 + h) * USTR + q] = U[i];
        }
        if (lane < 8) sTh[g * 8 + lane] = th[lane];
        for (int i = lane; i < 128; i += 32) sCb[g * 128 + i] = cb[i];
    }
    for (int i = lane; i < 16 * HSTR; i += 32) { hbuf[i] = 0.0f; cbuf[i] = 0.0f; }
    __syncthreads();

    const int chunk   = blockIdx.x;     // 0..15
    const int rowbase = chunk * 16;     // batch rows [rowbase, rowbase+16)

    for (int t = 0; t < T_STEPS; ++t) {
        // ---- z = h @ Wh.T + zx[t]  ->  zbuf (16 x 32) ----
        {
            v8f c0 = {}, c1 = {}, d0 = {}, d1 = {};
            #pragma unroll 4
            for (int kk = 0; kk < 128; kk += 8) {
                int ka = kk + 2 * half;
                int kb = kk + 4 + 2 * half;
                v2f a0  = *(const v2f*)(hbuf + lm * HSTR + ka);
                v2f a1  = *(const v2f*)(hbuf + lm * HSTR + kb);
                v2f b00 = *(const v2f*)(sWh + lm * HSTR + ka);
                v2f b10 = *(const v2f*)(sWh + (16 + lm) * HSTR + ka);
                v2f b01 = *(const v2f*)(sWh + lm * HSTR + kb);
                v2f b11 = *(const v2f*)(sWh + (16 + lm) * HSTR + kb);
                c0 = __builtin_amdgcn_wmma_f32_16x16x4_f32(false, a0, false, b00, (short)0, c0, false, false);
                c1 = __builtin_amdgcn_wmma_f32_16x16x4_f32(false, a0, false, b10, (short)0, c1, false, false);
                d0 = __builtin_amdgcn_wmma_f32_16x16x4_f32(false, a1, false, b01, (short)0, d0, false, false);
                d1 = __builtin_amdgcn_wmma_f32_16x16x4_f32(false, a1, false, b11, (short)0, d1, false, false);
            }
            c0 = c0 + d0;
            c1 = c1 + d1;
            #pragma unroll
            for (int r = 0; r < 8; ++r) {
                int m = r + 8 * half;
                const float* zrow = zx + ((size_t)t * BATCH + rowbase + m) * 32;
                zbuf[m * ZSTR + lm]      = c0[r] + zrow[lm];
                zbuf[m * ZSTR + 16 + lm] = c1[r] + zrow[16 + lm];
            }
        }

        // ---- qgate: 64 (row, gate) pairs, 2 per lane ----
        #pragma unroll
        for (int pp = 0; pp < 2; ++pp) {
            int p = lane + 32 * pp;
            int b = p >> 2, g = p & 3;
            float cs[8];
            #pragma unroll
            for (int q = 0; q < 8; ++q)
                cs[q] = __cosf(zbuf[b * ZSTR + g * 8 + q] + sTh[g * 8 + q]);
            float tail = 1.0f;
            #pragma unroll
            for (int q = 1; q < 8; ++q) tail *= cs[q];
            qbuf[b * ZSTR + g * 8 + 0] = tail;        // <Z_0> = prod_{q>=1} cos
            float cum = cs[0];
            #pragma unroll
            for (int q = 1; q < 8; ++q) {             // <Z_w> = cumprod up to w
                cum *= cs[q];
                qbuf[b * ZSTR + g * 8 + q] = cum;
            }
        }

        // ---- per gate: a_g = qout_g @ Ug.T + cb, fused LSTM elementwise ----
        for (int g = 0; g < 4; ++g) {
            for (int ng = 0; ng < 2; ++ng) {          // 2 groups of 4 N-tiles
                v8f acc[4];
                #pragma unroll
                for (int j = 0; j < 4; ++j) {
                    float bias = sCb[g * 128 + ng * 64 + j * 16 + lm];
                    #pragma unroll
                    for (int r = 0; r < 8; ++r) acc[j][r] = bias;
                }
                #pragma unroll
                for (int kk = 0; kk < 8; kk += 4) {
                    int k0 = kk + 2 * half;
                    v2f a = *(const v2f*)(qbuf + lm * ZSTR + g * 8 + k0);
                    #pragma unroll
                    for (int j = 0; j < 4; ++j) {
                        int n = ng * 64 + j * 16 + lm;
                        v2f b = *(const v2f*)(sU + (g * 128 + n) * USTR + k0);
                        acc[j] = __builtin_amdgcn_wmma_f32_16x16x4_f32(false, a, false, b,
                                                                       (short)0, acc[j], false, false);
                    }
                }
                #pragma unroll
                for (int j = 0; j < 4; ++j) {
                    #pragma unroll
                    for (int r = 0; r < 8; ++r) {
                        int m = r + 8 * half, n = ng * 64 + j * 16 + lm;
                        int idx = m * HSTR + n;
                        float av = acc[j][r];
                        if (g == 0) {                        // forget
                            cbuf[idx] = cbuf[idx] * sigmoidf_(av);
                        } else if (g == 1) {                 // input
                            ibuf[idx] = sigmoidf_(av);
                        } else if (g == 2) {                 // candidate
                            cbuf[idx] += ibuf[idx] * tanhf_(av);
                        } else {                             // output
                            float hv = sigmoidf_(av) * tanhf_(cbuf[idx]);
                            hbuf[idx] = hv;
                            outs[((size_t)t * BATCH + rowbase + m) * HID + n] = hv;
                        }
                    }
                }
            }
        }
    }

    // ---- final h, c ----
    for (int i = lane; i < 16 * 128; i += 32) {
        int m = i >> 7, n = i & 127;
        hx[(size_t)(rowbase + m) * HID + n] = hbuf[m * HSTR + n];
        cx[(size_t)(rowbase + m) * HID + n] = cbuf[m * HSTR + n];
    }
}

// ---------------------------------------------------------------------------
extern "C" void kernel_launch(void* const* d_in, const int* in_sizes, int n_in,
                              void* d_out, int out_size, void* d_ws, size_t ws_size,
                              hipStream_t stream) {
    (void)in_sizes; (void)n_in; (void)out_size; (void)ws_size;

    const float* X   = (const float*)d_in[0];
    const float* Wf  = (const float*)d_in[1];
    const float* bf  = (const float*)d_in[2];
    const float* thf = (const float*)d_in[3];
    const float* Uf  = (const float*)d_in[4];
    const float* cf  = (const float*)d_in[5];
    const float* Wi  = (const float*)d_in[6];
    const float* bi  = (const float*)d_in[7];
    const float* thi = (const float*)d_in[8];
    const float* Ui  = (const float*)d_in[9];
    const float* ci  = (const float*)d_in[10];
    const float* Wg  = (const float*)d_in[11];
    const float* bg  = (const float*)d_in[12];
    const float* thg = (const float*)d_in[13];
    const float* Ug  = (const float*)d_in[14];
    const float* cg  = (const float*)d_in[15];
    const float* Wo  = (const float*)d_in[16];
    const float* bo  = (const float*)d_in[17];
    const float* tho = (const float*)d_in[18];
    const float* Uo  = (const float*)d_in[19];
    const float* co  = (const float*)d_in[20];

    float* outs = (float*)d_out;                            // (T, B, H)
    float* hx   = outs + (size_t)T_STEPS * BATCH * HID;     // (B, H)
    float* cx   = hx + (size_t)BATCH * HID;                 // (B, H)
    float* zxws = (float*)d_ws;                             // (T*B, 32) = 16 MB

    // Pass 1: parallel input projection (x-part of all 4 gates + bias).
    qlstm_zx_kernel<<<(T_STEPS * BATCH / 16) / 4, 128, 0, stream>>>(
        X, Wf, bf, Wi, bi, Wg, bg, Wo, bo, zxws);

    // Pass 2: sequential scan, 16 independent single-wave workgroups.
    qlstm_scan_kernel<<<16, 32, 0, stream>>>(
        zxws, Wf, thf, Uf, cf, Wi, thi, Ui, ci,
        Wg, thg, Ug, cg, Wo, tho, Uo, co, outs, hx, cx);
}